// ImgtoClass_Metric_86663850099093
// MI455X (gfx1250) — compile-verified
//
#include <hip/hip_runtime.h>
#include <math.h>
#include <stdint.h>

// Problem constants (from the reference)
static constexpr int kC   = 64;    // channels (K dim)
static constexpr int kHW  = 441;   // 21*21 descriptors per image
static constexpr int kHWP = 448;   // padded to 28 tiles of 16
static constexpr int kNT  = 28;    // 16-wide tiles per spatial dim

// LDS row stride for the staged B operand: 64 data floats + 4 pad floats.
// stride mod 64 banks == 4, so lane l of each 16-lane half covers banks
// {4l, 4l+1} (+2 for the upper half) -> a wave32 ds_load_b64 hits all 64
// banks exactly once (conflict-free).
static constexpr int kLdsStride = 68;                   // floats
static constexpr int kSmemBytes = kHWP * kLdsStride * 4; // 121,856 B < 320 KB

typedef float v2f __attribute__((ext_vector_type(2)));
typedef float v8f __attribute__((ext_vector_type(8)));

// ---------------------------------------------------------------------------
// Kernel 1: fold mask / L2-norm into each descriptor; transpose from
// (b, c, pos) to (b, pos_padded, c) row-major so both WMMA A and B fragments
// become a single b64 load per lane. Padded positions written as exact zeros.
// ---------------------------------------------------------------------------
__global__ void normalize_kernel(const float* __restrict__ x,
                                 const float* __restrict__ mask,
                                 float* __restrict__ out, int B) {
    int idx = blockIdx.x * blockDim.x + threadIdx.x;
    int total = B * kHWP;
    if (idx >= total) return;
    int b = idx / kHWP;
    int p = idx - b * kHWP;
    float* o = out + ((size_t)b * kHWP + p) * kC;
    if (p >= kHW) {
#pragma unroll
        for (int c = 0; c < kC; ++c) o[c] = 0.0f;
        return;
    }
    const float* xb = x + (size_t)b * kC * kHW + p;
    float s = 0.0f;
#pragma unroll
    for (int c = 0; c < kC; ++c) { float v = xb[c * kHW]; s += v * v; }
    float f = mask[(size_t)b * kHW + p] / sqrtf(s);
#pragma unroll
    for (int c = 0; c < kC; ++c) o[c] = xb[c * kHW] * f;
}

// Branchless top-3 insert: 3 max + 2 min VALU ops.
__device__ __forceinline__ void top3_insert(float v, float& t0, float& t1, float& t2) {
    float a0 = fmaxf(t0, v);
    float b0 = fminf(t0, v);
    float a1 = fmaxf(t1, b0);
    float b1 = fminf(t1, b0);
    t2 = fmaxf(t2, b1);
    t1 = a1;
    t0 = a0;
}

// ---------------------------------------------------------------------------
// Kernel 2: one workgroup per (a,b) pair. The support operand S_b is staged
// into LDS once per workgroup via global_load_async_to_lds_b128 (ASYNCcnt),
// with a bank-conflict-free padded stride. Each of the 8 waves owns M-tiles
// wave, wave+8, ...; the 16x64 A strip stays resident in VGPRs and two
// independent v_wmma_f32_16x16x4_f32 accumulator chains (N-tiles nt, nt+1)
// are interleaved per pass. Per-row top-3 + sum fused in registers/shuffles.
// ---------------------------------------------------------------------------
__global__ __launch_bounds__(256)
void simtopk_kernel(const float* __restrict__ Q, const float* __restrict__ S,
                    float* __restrict__ out, int B2) {
    extern __shared__ float smem[];   // kHWP rows x kLdsStride floats

    const int pair = blockIdx.x;
    const int a = pair / B2;
    const int b = pair - a * B2;
    const float* Qa = Q + (size_t)a * kHWP * kC;
    const float* Sb = S + (size_t)b * kHWP * kC;

    const int tid  = threadIdx.x;
    const int wave = tid >> 5;
    const int lane = tid & 31;
    const int l16  = lane & 15;
    const int kofs = (lane >> 4) * 2;  // channel offset within a K=4 step

    __shared__ float wsum[8];

    // ---- Stage S_b into LDS: 448 rows x 256 B = 7168 chunks of 16 B.
    // 256 threads x 28 async b128 loads, global -> LDS without touching VGPRs.
#pragma unroll
    for (int i = 0; i < 28; ++i) {
        int ch   = tid + i * 256;      // 0..7167
        int row  = ch >> 4;            // source row (64 floats)
        int part = ch & 15;            // 16-byte chunk within row
        unsigned lds  = (unsigned)(uintptr_t)(&smem[row * kLdsStride]) + part * 16;
        unsigned goff = row * 256 + part * 16;
        asm volatile("global_load_async_to_lds_b128 %0, %1, %2"
                     :: "v"(lds), "v"(goff), "s"(Sb)
                     : "memory");
    }
#if __has_builtin(__builtin_amdgcn_s_wait_asynccnt)
    __builtin_amdgcn_s_wait_asynccnt(0);
#else
    asm volatile("s_wait_asynccnt 0x0" ::: "memory");
#endif
    __syncthreads();

    float acc = 0.0f;

    for (int mt = wave; mt < kNT; mt += 8) {
        // Resident A fragments: 16 K-steps x v2f (channels 4k+kofs, 4k+kofs+1)
        v2f afrag[16];
        const float* arow = Qa + (size_t)(mt * 16 + l16) * kC + kofs;
#pragma unroll
        for (int k = 0; k < 16; ++k)
            afrag[k] = *(const v2f*)(arow + k * 4);

        float t0[8], t1[8], t2[8];
#pragma unroll
        for (int r = 0; r < 8; ++r) { t0[r] = -INFINITY; t1[r] = -INFINITY; t2[r] = -INFINITY; }

        for (int nt = 0; nt < kNT; nt += 2) {
            // B fragments from LDS (conflict-free ds_load_b64)
            const float* b0p = smem + (nt * 16 + l16) * kLdsStride + kofs;
            const float* b1p = b0p + 16 * kLdsStride;
            v8f c0 = {0.f, 0.f, 0.f, 0.f, 0.f, 0.f, 0.f, 0.f};
            v8f c1 = {0.f, 0.f, 0.f, 0.f, 0.f, 0.f, 0.f, 0.f};
#pragma unroll
            for (int k = 0; k < 16; ++k) {
                v2f b0 = *(const v2f*)(b0p + 4 * k);
                v2f b1 = *(const v2f*)(b1p + 4 * k);
                // Two independent fp32 WMMA chains share the A fragment.
                c0 = __builtin_amdgcn_wmma_f32_16x16x4_f32(
                         false, afrag[k], false, b0, (short)0, c0, false, false);
                c1 = __builtin_amdgcn_wmma_f32_16x16x4_f32(
                         false, afrag[k], false, b1, (short)0, c1, false, false);
            }
            // C layout: VGPR r, lane l -> row = r + 8*(l/16), col = l%16.
            // col0 <= 26*16+15 = 431 < 441: always valid.
            const int col1 = nt * 16 + 16 + l16;
            const bool valid1 = col1 < kHW;  // mask padded support columns
#pragma unroll
            for (int r = 0; r < 8; ++r) {
                top3_insert(c0[r], t0[r], t1[r], t2[r]);
                float v1 = valid1 ? c1[r] : -INFINITY;
                top3_insert(v1, t0[r], t1[r], t2[r]);
            }
        }

        // Merge per-lane top-3 sets across the 16 lanes of each half-wave
        // (xor masks 1,2,4,8 never cross the 16-lane boundary).
#pragma unroll
        for (int m = 1; m <= 8; m <<= 1) {
#pragma unroll
            for (int r = 0; r < 8; ++r) {
                float p0 = __shfl_xor(t0[r], m, 32);
                float p1 = __shfl_xor(t1[r], m, 32);
                float p2 = __shfl_xor(t2[r], m, 32);
                top3_insert(p0, t0[r], t1[r], t2[r]);
                top3_insert(p1, t0[r], t1[r], t2[r]);
                top3_insert(p2, t0[r], t1[r], t2[r]);
            }
        }
        float p = 0.0f;
#pragma unroll
        for (int r = 0; r < 8; ++r) p += t0[r] + t1[r] + t2[r];
        // lanes 0-15 hold rows mt*16..+7, lanes 16-31 hold rows +8..+15
        acc += __shfl(p, 0, 32) + __shfl(p, 16, 32);
    }

    if (lane == 0) wsum[wave] = acc;
    __syncthreads();
    if (tid == 0) {
        float s = 0.0f;
#pragma unroll
        for (int w = 0; w < 8; ++w) s += wsum[w];
        out[pair] = s;
    }
}

// ---------------------------------------------------------------------------
extern "C" void kernel_launch(void* const* d_in, const int* in_sizes, int n_in,
                              void* d_out, int out_size, void* d_ws, size_t ws_size,
                              hipStream_t stream) {
    const float* x1 = (const float*)d_in[0];
    const float* x2 = (const float*)d_in[1];
    const float* m1 = (const float*)d_in[2];
    const float* m2 = (const float*)d_in[3];

    const int B1 = in_sizes[0] / (kC * kHW);
    const int B2 = in_sizes[1] / (kC * kHW);

    float* Qws = (float*)d_ws;                       // B1 * 448 * 64 floats
    float* Sws = Qws + (size_t)B1 * kHWP * kC;       // B2 * 448 * 64 floats

    const int tq = B1 * kHWP;
    const int ts = B2 * kHWP;
    normalize_kernel<<<(tq + 255) / 256, 256, 0, stream>>>(x1, m1, Qws, B1);
    normalize_kernel<<<(ts + 255) / 256, 256, 0, stream>>>(x2, m2, Sws, B2);

    simtopk_kernel<<<B1 * B2, 256, kSmemBytes, stream>>>(Qws, Sws, (float*)d_out, B2);
}